// QuaternionBatchNorm_20787641713084
// MI455X (gfx1250) — compile-verified
//
#include <hip/hip_runtime.h>
#include <math.h>

// QuaternionBatchNorm (whitening BN) for MI455X / gfx1250.
// Phase 1a: coalesced streaming reduction of per-feature 1st/2nd moments.
// Phase 1b: per-feature Cholesky + fallbacks, L^-1, fold gamma/mean/beta
//           into a single 4x4 matrix M_f and bias b'_f.
// Phase 2 : out = M_f x + b' via chained V_WMMA_F32_16X16X4_F32
//           (16 batches x 4 features per wave tile, block-diagonal B).

typedef float v2f __attribute__((ext_vector_type(2)));
typedef float v8f __attribute__((ext_vector_type(8)));

#define NPART 64          // batch partitions for the reduction
#define QBN_EPS 1e-5f

// ---------------------------------------------------------------------------
// Phase 1a: partial sums. grid = (F/256, NPART), block = 256.
// Thread t of block (bx, k) owns feature f = bx*256 + t and batches
// b = k, k+NPART, ... Consecutive threads read consecutive float4s (4 KB
// contiguous per iteration) -> fully coalesced.
// Partials layout: part[(c*NPART + k)*F + f], c = 0..13
//   c 0..3  : sum(x_i)
//   c 4..13 : sum(x_i * x_j) in tril order (00,10,11,20,21,22,30,31,32,33)
// ---------------------------------------------------------------------------
__global__ void __launch_bounds__(256)
qbn_partials(const float* __restrict__ x, float* __restrict__ part,
             int B, int F) {
  const int f = blockIdx.x * blockDim.x + threadIdx.x;
  const int k = blockIdx.y;

  float s0 = 0.f, s1 = 0.f, s2 = 0.f, s3 = 0.f;
  float p00 = 0.f, p10 = 0.f, p11 = 0.f, p20 = 0.f, p21 = 0.f;
  float p22 = 0.f, p30 = 0.f, p31 = 0.f, p32 = 0.f, p33 = 0.f;

  for (int b = k; b < B; b += NPART) {
    const float4 v = *(const float4*)(x + ((size_t)b * F + f) * 4);
    s0 += v.x; s1 += v.y; s2 += v.z; s3 += v.w;
    p00 = fmaf(v.x, v.x, p00);
    p10 = fmaf(v.y, v.x, p10);
    p11 = fmaf(v.y, v.y, p11);
    p20 = fmaf(v.z, v.x, p20);
    p21 = fmaf(v.z, v.y, p21);
    p22 = fmaf(v.z, v.z, p22);
    p30 = fmaf(v.w, v.x, p30);
    p31 = fmaf(v.w, v.y, p31);
    p32 = fmaf(v.w, v.z, p32);
    p33 = fmaf(v.w, v.w, p33);
  }

  const size_t stride = (size_t)NPART * F;
  size_t base = (size_t)k * F + f;
  float vals[14] = {s0, s1, s2, s3, p00, p10, p11, p20, p21,
                    p22, p30, p31, p32, p33};
#pragma unroll
  for (int c = 0; c < 14; ++c) part[base + (size_t)c * stride] = vals[c];
}

// ---------------------------------------------------------------------------
// 4x4 Cholesky on tril-packed symmetric input c[10] (+shift*I).
// Returns false if any entry is NaN (mirrors jnp.linalg.cholesky NaN-on-fail).
// ---------------------------------------------------------------------------
__device__ inline bool chol4(const float c[10], float shift, float L[10]) {
  float a00 = c[0] + shift, a11 = c[2] + shift;
  float a22 = c[5] + shift, a33 = c[9] + shift;
  float L00 = sqrtf(a00);
  float L10 = c[1] / L00, L20 = c[3] / L00, L30 = c[6] / L00;
  float L11 = sqrtf(a11 - L10 * L10);
  float L21 = (c[4] - L20 * L10) / L11;
  float L31 = (c[7] - L30 * L10) / L11;
  float L22 = sqrtf(a22 - L20 * L20 - L21 * L21);
  float L32 = (c[8] - L30 * L20 - L31 * L21) / L22;
  float L33 = sqrtf(a33 - L30 * L30 - L31 * L31 - L32 * L32);
  L[0] = L00; L[1] = L10; L[2] = L11; L[3] = L20; L[4] = L21;
  L[5] = L22; L[6] = L30; L[7] = L31; L[8] = L32; L[9] = L33;
  bool bad = false;
#pragma unroll
  for (int t = 0; t < 10; ++t) bad |= (L[t] != L[t]);
  return !bad;
}

// ---------------------------------------------------------------------------
// Phase 1b: per-feature finalize. One thread per feature.
// Writes Mout[f*16 + i*4 + j] = (Gamma_sym @ L^-1)[i][j] and
//        bout[f*4 + i]       = beta[i] - sum_j M[i][j]*mean[j]
// ---------------------------------------------------------------------------
__global__ void __launch_bounds__(256)
qbn_finalize(const float* __restrict__ part, const float* __restrict__ gamma,
             const float* __restrict__ beta, float* __restrict__ Mout,
             float* __restrict__ bout, int B, int F) {
  const int f = blockIdx.x * blockDim.x + threadIdx.x;
  if (f >= F) return;

  float acc[14];
#pragma unroll
  for (int c = 0; c < 14; ++c) {
    float s = 0.f;
    const float* p = part + ((size_t)c * NPART) * F + f;
    for (int k = 0; k < NPART; ++k) s += p[(size_t)k * F];
    acc[c] = s;
  }

  const float invB = 1.0f / (float)B;
  float m[4] = {acc[0] * invB, acc[1] * invB, acc[2] * invB, acc[3] * invB};
  // covariance in tril order: E[x x^T] - mean mean^T
  float cov[10];
  cov[0] = acc[4] * invB - m[0] * m[0];
  cov[1] = acc[5] * invB - m[1] * m[0];
  cov[2] = acc[6] * invB - m[1] * m[1];
  cov[3] = acc[7] * invB - m[2] * m[0];
  cov[4] = acc[8] * invB - m[2] * m[1];
  cov[5] = acc[9] * invB - m[2] * m[2];
  cov[6] = acc[10] * invB - m[3] * m[0];
  cov[7] = acc[11] * invB - m[3] * m[1];
  cov[8] = acc[12] * invB - m[3] * m[2];
  cov[9] = acc[13] * invB - m[3] * m[3];

  // Cholesky with fallback ladder (eps -> 0.1 -> scaled identity)
  float L[10];
  if (!chol4(cov, QBN_EPS, L)) {
    float L2[10];
    if (chol4(cov, 0.1f, L2)) {
#pragma unroll
      for (int t = 0; t < 10; ++t) L[t] = L2[t];
    } else {
      float dm = 0.25f * (fabsf(cov[0]) + fabsf(cov[2]) + fabsf(cov[5]) +
                          fabsf(cov[9]));
      float sc = sqrtf(fmaxf(dm, 1.0f));
      L[0] = sc; L[1] = 0.f; L[2] = sc; L[3] = 0.f; L[4] = 0.f;
      L[5] = sc; L[6] = 0.f; L[7] = 0.f; L[8] = 0.f; L[9] = sc;
    }
  }

  // L^-1 by forward substitution (lower triangular)
  float Li[4][4] = {};
  Li[0][0] = 1.0f / L[0];
  Li[1][1] = 1.0f / L[2];
  Li[2][2] = 1.0f / L[5];
  Li[3][3] = 1.0f / L[9];
  Li[1][0] = -(L[1] * Li[0][0]) * Li[1][1];
  Li[2][0] = -(L[3] * Li[0][0] + L[4] * Li[1][0]) * Li[2][2];
  Li[2][1] = -(L[4] * Li[1][1]) * Li[2][2];
  Li[3][0] = -(L[6] * Li[0][0] + L[7] * Li[1][0] + L[8] * Li[2][0]) * Li[3][3];
  Li[3][1] = -(L[7] * Li[1][1] + L[8] * Li[2][1]) * Li[3][3];
  Li[3][2] = -(L[8] * Li[2][2]) * Li[3][3];

  // gamma_sym from tril order (0,0)(1,0)(1,1)(2,0)(2,1)(2,2)(3,0)(3,1)(3,2)(3,3)
  float g[10];
#pragma unroll
  for (int t = 0; t < 10; ++t) g[t] = gamma[(size_t)f * 10 + t];
  float G[4][4];
  G[0][0] = g[0];
  G[1][0] = G[0][1] = g[1];
  G[1][1] = g[2];
  G[2][0] = G[0][2] = g[3];
  G[2][1] = G[1][2] = g[4];
  G[2][2] = g[5];
  G[3][0] = G[0][3] = g[6];
  G[3][1] = G[1][3] = g[7];
  G[3][2] = G[2][3] = g[8];
  G[3][3] = g[9];

  // M = G @ L^-1 ; b' = beta - M @ mean
  float Mm[4][4];
#pragma unroll
  for (int i = 0; i < 4; ++i)
#pragma unroll
    for (int j = 0; j < 4; ++j) {
      float s = 0.f;
#pragma unroll
      for (int kk = 0; kk < 4; ++kk) s = fmaf(G[i][kk], Li[kk][j], s);
      Mm[i][j] = s;
    }

#pragma unroll
  for (int i = 0; i < 4; ++i) {
    float s = beta[(size_t)f * 4 + i];
#pragma unroll
    for (int j = 0; j < 4; ++j) s = fmaf(-Mm[i][j], m[j], s);
    bout[(size_t)f * 4 + i] = s;
#pragma unroll
    for (int j = 0; j < 4; ++j) Mout[(size_t)f * 16 + i * 4 + j] = Mm[i][j];
  }
}

// ---------------------------------------------------------------------------
// Phase 2: out = M_f x + b' via V_WMMA_F32_16X16X4_F32.
// Tile = 16 batches x 4 features (16 output cols = 64 contiguous bytes/row).
// D[16x16] = sum_c A_c[16x4] * B_c[4x16] + C, B_c block-diagonal slices.
// A layout (ISA 32-bit 16x4): lane L holds (M=L&15, K=2*(L>>4)+v), v=0..1.
// B layout (row per K striped over lanes): lane L, vgpr v -> (K=2*(L>>4)+v,
// N=L&15). C/D: lane L vgpr v -> (M=v+8*(L>>4), N=L&15).
// ---------------------------------------------------------------------------
__global__ void __launch_bounds__(256)
qbn_apply(const float* __restrict__ x, const float* __restrict__ Mws,
          const float* __restrict__ bws, float* __restrict__ out,
          int B, int F) {
  const int wave = threadIdx.x >> 5;
  const int lane = threadIdx.x & 31;
  const int FG = F >> 2;  // feature groups of 4

  const int tile = blockIdx.x * (blockDim.x >> 5) + wave;
  const int fg = tile % FG;
  const int bt = tile / FG;
  const int f0 = fg << 2;
  const int b0 = bt << 4;

  const int n = lane & 15;     // output column: 4*(feature-in-group) + dim
  const int half = lane >> 4;  // K half (covers K pair 2*half, 2*half+1)
  const int fsel = n >> 2;     // which feature this column belongs to
  const int i = n & 3;         // output dim within feature
  const int f = f0 + fsel;

  // A fragments: lane's row is b0+n; 4 K-chunks = 4 float2 loads.
  const float* xr = x + ((size_t)(b0 + n) * F + f0) * 4 + 2 * half;
  const v2f a0 = *(const v2f*)(xr);
  const v2f a1 = *(const v2f*)(xr + 4);
  const v2f a2 = *(const v2f*)(xr + 8);
  const v2f a3 = *(const v2f*)(xr + 12);

  // B fragments: column n needs M_f[i][K]; zero outside its diagonal block.
  const v2f mv = *(const v2f*)(Mws + (size_t)f * 16 + i * 4 + 2 * half);
  const v2f z = {0.f, 0.f};
  const v2f bf0 = (fsel == 0) ? mv : z;
  const v2f bf1 = (fsel == 1) ? mv : z;
  const v2f bf2 = (fsel == 2) ? mv : z;
  const v2f bf3 = (fsel == 3) ? mv : z;

  // C preloaded with folded bias (constant down each output column).
  const float bv = bws[(size_t)f * 4 + i];
  v8f acc = {bv, bv, bv, bv, bv, bv, bv, bv};

  acc = __builtin_amdgcn_wmma_f32_16x16x4_f32(false, a0, false, bf0,
                                              (short)0, acc, false, false);
  acc = __builtin_amdgcn_wmma_f32_16x16x4_f32(false, a1, false, bf1,
                                              (short)0, acc, false, false);
  acc = __builtin_amdgcn_wmma_f32_16x16x4_f32(false, a2, false, bf2,
                                              (short)0, acc, false, false);
  acc = __builtin_amdgcn_wmma_f32_16x16x4_f32(false, a3, false, bf3,
                                              (short)0, acc, false, false);

  // D: vgpr v -> row b0 + v + 8*half, col n (16 contiguous floats per row).
  float* orow = out + ((size_t)(b0 + 8 * half) * F + f0) * 4 + n;
  const size_t rs = (size_t)F * 4;
#pragma unroll
  for (int v = 0; v < 8; ++v) orow[(size_t)v * rs] = acc[v];
}

// ---------------------------------------------------------------------------
extern "C" void kernel_launch(void* const* d_in, const int* in_sizes, int n_in,
                              void* d_out, int out_size, void* d_ws,
                              size_t ws_size, hipStream_t stream) {
  (void)n_in; (void)out_size; (void)ws_size;
  const float* x = (const float*)d_in[0];      // [B, F, 4]
  const float* gamma = (const float*)d_in[1];  // [F, 10]
  const float* beta = (const float*)d_in[2];   // [F, 4]
  float* out = (float*)d_out;                  // [B, F, 4]

  const int F = in_sizes[2] / 4;               // beta is [F,4]
  const int B = in_sizes[0] / (F * 4);

  // Workspace: partials [14][NPART][F], then M [F][16], then b' [F][4]
  float* part = (float*)d_ws;
  float* Mws = part + (size_t)14 * NPART * F;
  float* bws = Mws + (size_t)F * 16;

  // Phase 1a: moment partials
  dim3 g1(F / 256, NPART);
  qbn_partials<<<g1, 256, 0, stream>>>(x, part, B, F);

  // Phase 1b: per-feature whitening matrix + bias fold
  qbn_finalize<<<(F + 255) / 256, 256, 0, stream>>>(part, gamma, beta, Mws,
                                                    bws, B, F);

  // Phase 2: WMMA transform. One 16x4 tile per wave, 8 waves per block.
  const int numTiles = (B / 16) * (F / 4);
  qbn_apply<<<numTiles / 8, 256, 0, stream>>>(x, Mws, bws, out, B, F);
}